// WPDCAxisAngleLoss_37366215475314
// MI455X (gfx1250) — compile-verified
//
#include <hip/hip_runtime.h>
#include <math.h>

// ---------------------------------------------------------------------------
// WPDC axis-angle loss, reduced to closed form via per-sample base moments.
//   base[n,l] = u[l] + w_shp[l,:]·a_shp[n] + w_exp[l,:]·a_exp[n]
//   For candidate c: dist^2 = tr(MᵀM·B) + 2 dtᵀ M s1 + NV·|dt|²
//   with B[n]=Σ_v b bᵀ, s1[n]=Σ_v b  (b = base triple per vertex).
// Heavy kernel: fp32 WMMA GEMM (W 48x52 tile in LDS  ×  Acoef 52x128) fused
// with moment accumulation. W is streamed from HBM exactly once.
// ---------------------------------------------------------------------------

#define EPSV 1e-8f
#define NSAMP 128
#define NVERT 53215
#define LTOT (3 * NVERT)
#define KDIM 52              // 1 (u) + 40 (shp) + 10 (exp) + 1 pad
#define CHUNK_ROWS 48        // 16 vertices per chunk
#define NCHUNK ((NVERT + 15) / 16)

typedef __attribute__((ext_vector_type(2))) float v2f;
typedef __attribute__((ext_vector_type(8))) float v8f;

// workspace layout in floats
#define WS_ACOEF 0
#define WS_AA7   (WS_ACOEF + NSAMP * KDIM)   // stride 8 per sample
#define WS_PG    (WS_AA7 + NSAMP * 8)        // stride 12
#define WS_OFF   (WS_PG + NSAMP * 12)        // stride 4
#define WS_MOM   (WS_OFF + NSAMP * 4)        // stride 12 (9 used)

// ------------------------------- prep --------------------------------------
__global__ void __launch_bounds__(128)
wpdc_prep_kernel(const float* __restrict__ target,
                 const float* __restrict__ param_mean,
                 const float* __restrict__ param_std,
                 float* __restrict__ ws) {
  const int n = threadIdx.x;
  if (n >= NSAMP) return;

  float td[62];
#pragma unroll
  for (int i = 0; i < 62; ++i)
    td[i] = param_std[i] * target[n * 62 + i] + param_mean[i];

  // Acoef row (WMMA B operand source): [1, a_shp(40), a_exp(10), 0]
  float* ac = ws + WS_ACOEF + n * KDIM;
  ac[0] = 1.0f;
#pragma unroll
  for (int k = 0; k < 40; ++k) ac[1 + k] = td[12 + k];
#pragma unroll
  for (int k = 0; k < 10; ++k) ac[41 + k] = td[52 + k];
  ac[51] = 0.0f;

  // pg (3x3) and offsetg (3)
  float* pg = ws + WS_PG + n * 12;
  float* og = ws + WS_OFF + n * 4;
#pragma unroll
  for (int r = 0; r < 3; ++r) {
#pragma unroll
    for (int c = 0; c < 3; ++c) pg[r * 3 + c] = td[r * 4 + c];
    og[r] = td[r * 4 + 3];
  }

  // aa7 = [s, axis*angle (3), t (3)]  (== pg_aa of the reference)
  float r1[3] = {td[0], td[1], td[2]};
  float r2[3] = {td[4], td[5], td[6]};
  float n1 = sqrtf(r1[0] * r1[0] + r1[1] * r1[1] + r1[2] * r1[2]);
  float n2 = sqrtf(r2[0] * r2[0] + r2[1] * r2[1] + r2[2] * r2[2]);
  float s = 0.5f * (n1 + n2);
  float i1 = 1.0f / (n1 + EPSV), i2 = 1.0f / (n2 + EPSV);
#pragma unroll
  for (int i = 0; i < 3; ++i) { r1[i] *= i1; r2[i] *= i2; }
  float r3[3] = {r1[1] * r2[2] - r1[2] * r2[1],
                 r1[2] * r2[0] - r1[0] * r2[2],
                 r1[0] * r2[1] - r1[1] * r2[0]};
  float n3 = sqrtf(r3[0] * r3[0] + r3[1] * r3[1] + r3[2] * r3[2]);
  float i3 = 1.0f / (n3 + EPSV);
#pragma unroll
  for (int i = 0; i < 3; ++i) r3[i] *= i3;
  float r2o[3] = {r3[1] * r1[2] - r3[2] * r1[1],
                  r3[2] * r1[0] - r3[0] * r1[2],
                  r3[0] * r1[1] - r3[1] * r1[0]};
  float tr = r1[0] + r2o[1] + r3[2];
  float carg = fminf(fmaxf(0.5f * (tr - 1.0f), -1.0f + 1e-7f), 1.0f - 1e-7f);
  float ang = acosf(carg);
  float v0 = r3[1] - r2o[2];
  float v1 = r1[2] - r3[0];
  float v2 = r2o[0] - r1[1];
  float dn = 1.0f / (2.0f * sinf(ang) + EPSV);

  float* a7 = ws + WS_AA7 + n * 8;
  a7[0] = s;
  a7[1] = v0 * dn * ang;
  a7[2] = v1 * dn * ang;
  a7[3] = v2 * dn * ang;
  a7[4] = td[3];
  a7[5] = td[7];
  a7[6] = td[11];
  a7[7] = 0.0f;

  // zero moment accumulators
  float* mom = ws + WS_MOM + n * 12;
#pragma unroll
  for (int i = 0; i < 12; ++i) mom[i] = 0.0f;
}

// --------------------- fused WMMA GEMM + moment kernel ---------------------
__global__ void __launch_bounds__(256)
wpdc_moment_kernel(const float* __restrict__ u,
                   const float* __restrict__ w_shp,
                   const float* __restrict__ w_exp,
                   float* __restrict__ ws) {
  __shared__ float sW[CHUNK_ROWS * KDIM];      // 48x52 W tile (9984 B)
  __shared__ float sBase[CHUNK_ROWS * NSAMP];  // 48x128 base tile (24576 B)

  const int tid = threadIdx.x;
  const int wave = tid >> 5;          // 8 waves, wave w -> samples [16w,16w+16)
  const int lane = tid & 31;
  const int lane_lo = lane & 15;
  const int koff = (lane >= 16) ? 2 : 0;
  const int rbase = (lane >= 16) ? 8 : 0;
  const int n_base = wave * 16;

  // Hoist WMMA B fragments (Acoef) for the whole kernel: 13 K-steps x v2f.
  v2f bfrag[13];
  {
    const float* ac = ws + WS_ACOEF + (n_base + lane_lo) * KDIM;
#pragma unroll
    for (int kk = 0; kk < 13; ++kk)
      bfrag[kk] = *(const v2f*)(ac + 4 * kk + koff);
  }

  float m0 = 0.f, m1 = 0.f, m2 = 0.f, m3 = 0.f, m4 = 0.f,
        m5 = 0.f, m6 = 0.f, m7 = 0.f, m8 = 0.f;
  const int nmom = tid & 127;  // sample this thread accumulates moments for
  const int slot = tid >> 7;   // 0/1: which 8 vertices of the chunk

  for (int chunk = blockIdx.x; chunk < NCHUNK; chunk += gridDim.x) {
    const int l0 = chunk * CHUNK_ROWS;
    __syncthreads();  // protect sW / sBase reuse across iterations

    // ---- cooperative LDS stage of W tile [l0, l0+48) x 52, zero padded ----
    for (int e = tid; e < CHUNK_ROWS; e += 256) {
      const int l = l0 + e;
      sW[e * KDIM + 0] = (l < LTOT) ? u[l] : 0.0f;
      sW[e * KDIM + 51] = 0.0f;
    }
    for (int f = tid; f < CHUNK_ROWS * 10; f += 256) {  // w_shp as float4
      const int row = f / 10, c4 = f - row * 10;
      const int l = l0 + row;
      float4 val = make_float4(0.f, 0.f, 0.f, 0.f);
      if (l < LTOT) val = ((const float4*)w_shp)[(size_t)l * 10 + c4];
      float* d = &sW[row * KDIM + 1 + 4 * c4];
      d[0] = val.x; d[1] = val.y; d[2] = val.z; d[3] = val.w;
    }
    for (int f = tid; f < CHUNK_ROWS * 5; f += 256) {   // w_exp as float2
      const int row = f / 5, c2 = f - row * 5;
      const int l = l0 + row;
      float2 val = make_float2(0.f, 0.f);
      if (l < LTOT) val = ((const float2*)w_exp)[(size_t)l * 5 + c2];
      float* d = &sW[row * KDIM + 41 + 2 * c2];
      d[0] = val.x; d[1] = val.y;
    }
    {  // hint the next chunk of the dominant stream into cache
      const int lnext = l0 + (int)gridDim.x * CHUNK_ROWS;
      if (tid == 0 && lnext < LTOT)
        __builtin_prefetch(&w_shp[(size_t)lnext * 40], 0, 1);
    }
    __syncthreads();

    // ---- WMMA: base tile (48 x 16 per wave) = W(48x52) x Acoef(52x16)ᵀ ----
    v8f D0 = {0.f, 0.f, 0.f, 0.f, 0.f, 0.f, 0.f, 0.f};
    v8f D1 = {0.f, 0.f, 0.f, 0.f, 0.f, 0.f, 0.f, 0.f};
    v8f D2 = {0.f, 0.f, 0.f, 0.f, 0.f, 0.f, 0.f, 0.f};
#pragma unroll
    for (int kk = 0; kk < 13; ++kk) {
      const int kc = 4 * kk + koff;
      const v2f b = bfrag[kk];
      const v2f a0 = *(const v2f*)&sW[(lane_lo) * KDIM + kc];
      const v2f a1 = *(const v2f*)&sW[(lane_lo + 16) * KDIM + kc];
      const v2f a2 = *(const v2f*)&sW[(lane_lo + 32) * KDIM + kc];
      D0 = __builtin_amdgcn_wmma_f32_16x16x4_f32(false, a0, false, b,
                                                 (short)0, D0, false, false);
      D1 = __builtin_amdgcn_wmma_f32_16x16x4_f32(false, a1, false, b,
                                                 (short)0, D1, false, false);
      D2 = __builtin_amdgcn_wmma_f32_16x16x4_f32(false, a2, false, b,
                                                 (short)0, D2, false, false);
    }

    // ---- spill base tile to LDS for vertex-triple (stride-3) access ----
    {
      const int col = n_base + lane_lo;
#pragma unroll
      for (int r = 0; r < 8; ++r) {
        sBase[(0 + rbase + r) * NSAMP + col] = D0[r];
        sBase[(16 + rbase + r) * NSAMP + col] = D1[r];
        sBase[(32 + rbase + r) * NSAMP + col] = D2[r];
      }
    }
    __syncthreads();

    // ---- moment accumulation: B += b bᵀ, s1 += b ----
    const int vbase = chunk * 16;
#pragma unroll
    for (int i = 0; i < 8; ++i) {
      const int vl = slot * 8 + i;
      if (vbase + vl < NVERT) {
        const float bx = sBase[(3 * vl + 0) * NSAMP + nmom];
        const float by = sBase[(3 * vl + 1) * NSAMP + nmom];
        const float bz = sBase[(3 * vl + 2) * NSAMP + nmom];
        m0 += bx * bx; m1 += bx * by; m2 += bx * bz;
        m3 += by * by; m4 += by * bz; m5 += bz * bz;
        m6 += bx;      m7 += by;      m8 += bz;
      }
    }
  }

  // ---- pairwise reduce (slot 1 -> slot 0) then global atomic add ----
  __syncthreads();
  if (slot == 1) {
    float* r = &sBase[nmom * 9];
    r[0] = m0; r[1] = m1; r[2] = m2; r[3] = m3; r[4] = m4;
    r[5] = m5; r[6] = m6; r[7] = m7; r[8] = m8;
  }
  __syncthreads();
  if (slot == 0) {
    const float* r = &sBase[nmom * 9];
    float* mom = ws + WS_MOM + nmom * 12;
    atomicAdd(&mom[0], m0 + r[0]);
    atomicAdd(&mom[1], m1 + r[1]);
    atomicAdd(&mom[2], m2 + r[2]);
    atomicAdd(&mom[3], m3 + r[3]);
    atomicAdd(&mom[4], m4 + r[4]);
    atomicAdd(&mom[5], m5 + r[5]);
    atomicAdd(&mom[6], m6 + r[6]);
    atomicAdd(&mom[7], m7 + r[7]);
    atomicAdd(&mom[8], m8 + r[8]);
  }
}

// ------------------------------ final loss ---------------------------------
__global__ void __launch_bounds__(128)
wpdc_final_kernel(const float* __restrict__ input,
                  const float* __restrict__ ws,
                  float* __restrict__ out) {
  __shared__ float red[128];
  const int n = threadIdx.x;

  const float* mom = ws + WS_MOM + n * 12;
  const float B00 = mom[0], B01 = mom[1], B02 = mom[2];
  const float B11 = mom[3], B12 = mom[4], B22 = mom[5];
  const float s1x = mom[6], s1y = mom[7], s1z = mom[8];
  const float* aa7 = ws + WS_AA7 + n * 8;
  const float* pg = ws + WS_PG + n * 12;
  const float* og = ws + WS_OFF + n * 4;
  const float* inp = input + n * 62;

  float wv[7];
#pragma unroll
  for (int c = 0; c < 7; ++c) {
    float cand[7];
#pragma unroll
    for (int j = 0; j < 7; ++j) cand[j] = (j == c) ? inp[c] : aa7[j];
    const float s = fabsf(cand[0]);
    // Rodrigues (matches reference: R = I + sinθ K + (1-cosθ) K²)
    const float ax = cand[1], ay = cand[2], az = cand[3];
    const float theta = sqrtf(ax * ax + ay * ay + az * az);
    const float inv = 1.0f / (theta + EPSV);
    const float kx = ax * inv, ky = ay * inv, kz = az * inv;
    const float st = sinf(theta), ct = cosf(theta);
    const float K[3][3] = {{0.f, -kz, ky}, {kz, 0.f, -kx}, {-ky, kx, 0.f}};
    float M[3][3];
#pragma unroll
    for (int i = 0; i < 3; ++i)
#pragma unroll
      for (int j = 0; j < 3; ++j) {
        float kk = 0.f;
#pragma unroll
        for (int m = 0; m < 3; ++m) kk += K[i][m] * K[m][j];
        const float R = ((i == j) ? 1.0f : 0.0f) + st * K[i][j] +
                        (1.0f - ct) * kk;
        M[i][j] = s * R - pg[i * 3 + j];
      }
    const float dt0 = cand[4] - og[0];
    const float dt1 = cand[5] - og[1];
    const float dt2 = cand[6] - og[2];
    // G = MᵀM (symmetric)
    float G[3][3];
#pragma unroll
    for (int j = 0; j < 3; ++j)
#pragma unroll
      for (int k = 0; k < 3; ++k)
        G[j][k] = M[0][j] * M[0][k] + M[1][j] * M[1][k] + M[2][j] * M[2][k];
    const float quad = G[0][0] * B00 + G[1][1] * B11 + G[2][2] * B22 +
                       2.0f * (G[0][1] * B01 + G[0][2] * B02 + G[1][2] * B12);
    float Ms1[3];
#pragma unroll
    for (int r = 0; r < 3; ++r)
      Ms1[r] = M[r][0] * s1x + M[r][1] * s1y + M[r][2] * s1z;
    const float lin = 2.0f * (dt0 * Ms1[0] + dt1 * Ms1[1] + dt2 * Ms1[2]);
    const float cst = (float)NVERT * (dt0 * dt0 + dt1 * dt1 + dt2 * dt2);
    const float d2 = quad + lin + cst;
    wv[c] = sqrtf(fmaxf(d2, 0.0f)) + 1e-6f;
  }

  float wmax = wv[0];
#pragma unroll
  for (int c = 1; c < 7; ++c) wmax = fmaxf(wmax, wv[c]);
  const float iwmax = 1.0f / wmax;

  float acc = 0.0f;
#pragma unroll
  for (int c = 0; c < 6; ++c) {  // channel 6 has weight 0
    const float w = wv[c] * iwmax;
    const float pa = (c == 0) ? fabsf(inp[0]) : inp[c];
    const float d = pa - aa7[c];
    acc += w * d * d;
  }

  red[n] = acc;
  __syncthreads();
#pragma unroll
  for (int sft = 64; sft > 0; sft >>= 1) {
    if (n < sft) red[n] += red[n + sft];
    __syncthreads();
  }
  if (n == 0) out[0] = red[0] / (128.0f * 7.0f);
}

// ------------------------------- launcher ----------------------------------
extern "C" void kernel_launch(void* const* d_in, const int* in_sizes, int n_in,
                              void* d_out, int out_size, void* d_ws,
                              size_t ws_size, hipStream_t stream) {
  (void)in_sizes; (void)n_in; (void)out_size; (void)ws_size;
  const float* input = (const float*)d_in[0];
  const float* target = (const float*)d_in[1];
  const float* u = (const float*)d_in[2];
  const float* w_shp = (const float*)d_in[3];
  const float* w_exp = (const float*)d_in[4];
  const float* pmean = (const float*)d_in[5];
  const float* pstd = (const float*)d_in[6];
  float* out = (float*)d_out;
  float* ws = (float*)d_ws;

  wpdc_prep_kernel<<<1, 128, 0, stream>>>(target, pmean, pstd, ws);
  wpdc_moment_kernel<<<dim3(1024), dim3(256), 0, stream>>>(u, w_shp, w_exp, ws);
  wpdc_final_kernel<<<1, 128, 0, stream>>>(input, ws, out);
}